// BasicCGInducer_59725815218445
// MI455X (gfx1250) — compile-verified
//
#include <hip/hip_runtime.h>
#include <hip/hip_bf16.h>
#include <math.h>
#include <stdint.h>

// ---------------------------------------------------------------------------
// CDNA5 (gfx1250) fp32 WMMA implementation of the compound-PCFG parameter
// head. Heavy GEMMs: V_WMMA_F32_16X16X4_F32 (wave32), tiles staged into LDS
// with GLOBAL_LOAD_ASYNC_TO_LDS (ASYNCcnt) where the copy is contiguous.
// ---------------------------------------------------------------------------

typedef __attribute__((ext_vector_type(2))) float v2f;
typedef __attribute__((ext_vector_type(8))) float v8f;

#define Q_ALL  2048
#define Q_RES  1024
#define Q_ARG  1024
#define DDIM   1024
#define VSZ    32000
#define NTOK   4096     // B*S = 32*128

// GEMM tiling: 8 waves/block, wave grid 4(M) x 2(N), each wave = 16x64 via 4 WMMA tiles
#define BM 64
#define BN 128
#define BK 32
#define LDP 36          // BK + 4 pad: row stride 144B -> 16B aligned (async b128)
                        // and even (8B-aligned float2 WMMA operand loads)

#define NSPLIT 10       // V-dimension split for the emit logsumexp pass
#define CHUNKS_PER_SPLIT 25   // 32000 / 128 / 10

__device__ __forceinline__ v8f v8f_zero() {
  v8f z = {0.f,0.f,0.f,0.f,0.f,0.f,0.f,0.f};
  return z;
}

__device__ __forceinline__ v8f wmma_f32(v2f a, v2f b, v8f c) {
  // V_WMMA_F32_16X16X4_F32: D = A(16x4) * B(4x16) + C(16x16)
  return __builtin_amdgcn_wmma_f32_16x16x4_f32(
      /*neg_a=*/false, a, /*neg_b=*/false, b,
      /*c_mod=*/(short)0, c, /*reuse_a=*/false, /*reuse_b=*/false);
}

// --- CDNA5 async global->LDS copies (tracked with ASYNCcnt) ----------------
__device__ __forceinline__ void async_lds_b128(void* lds, const void* gptr) {
  unsigned loff = (unsigned)(uintptr_t)lds;       // low 32 bits = LDS offset
  asm volatile("global_load_async_to_lds_b128 %0, %1, off"
               :: "v"(loff), "v"(gptr) : "memory");
}
__device__ __forceinline__ void async_lds_b32(void* lds, const void* gptr) {
  unsigned loff = (unsigned)(uintptr_t)lds;
  asm volatile("global_load_async_to_lds_b32 %0, %1, off"
               :: "v"(loff), "v"(gptr) : "memory");
}
__device__ __forceinline__ void wait_async0() {
  asm volatile("s_wait_asynccnt 0" ::: "memory");
}

// ---------------------------------------------------------------------------
// Generic fp32 WMMA GEMM: out = epi(A[MxK] @ W[KxN] + bias, relu?, +resid?)
// ---------------------------------------------------------------------------
__global__ __launch_bounds__(256)
void gemm_mlp(const float* __restrict__ A, const float* __restrict__ W,
              const float* __restrict__ bias, const float* __restrict__ resid,
              float* __restrict__ out, int M, int N, int K, int do_relu)
{
  __shared__ float As[BM][LDP];
  __shared__ float Bs[BN][LDP];           // stored transposed: Bs[n][k]
  const int t    = threadIdx.x;
  const int lane = t & 31;
  const int wave = t >> 5;
  const int wm   = wave >> 1;             // 0..3  (M sub-tile)
  const int wn   = wave & 1;              // 0..1  (N half)
  const int mbase = blockIdx.y * BM;
  const int nbase = blockIdx.x * BN;
  const int lm  = lane & 15;              // A row / B col within tile
  const int lk2 = (lane >> 4) * 2;        // K pair selected by lane half

  v8f acc[4];
  #pragma unroll
  for (int j = 0; j < 4; ++j) acc[j] = v8f_zero();

  for (int k0 = 0; k0 < K; k0 += BK) {
    __syncthreads();
    #pragma unroll
    for (int i = 0; i < 2; ++i) {                 // 64x32 A tile: async, 16B chunks
      int v = t + 256 * i;
      int r = v >> 3, c4 = (v & 7) * 4;
      async_lds_b128(&As[r][c4], &A[(mbase + r) * K + k0 + c4]);
    }
    #pragma unroll
    for (int i = 0; i < 16; ++i) {                // 32x128 W tile, transposed store
      int idx = t + 256 * i;
      int kr = idx >> 7, nc = idx & 127;
      Bs[nc][kr] = W[(k0 + kr) * N + nbase + nc];
    }
    wait_async0();
    __syncthreads();
    #pragma unroll
    for (int kk = 0; kk < BK; kk += 4) {
      v2f a = *(const v2f*)&As[wm * 16 + lm][kk + lk2];
      #pragma unroll
      for (int j = 0; j < 4; ++j) {
        v2f b = *(const v2f*)&Bs[wn * 64 + j * 16 + lm][kk + lk2];
        acc[j] = wmma_f32(a, b, acc[j]);
      }
    }
  }

  const int rbase = (lane >> 4) * 8;              // C/D: lanes>=16 hold rows 8..15
  #pragma unroll
  for (int j = 0; j < 4; ++j) {
    int gn = nbase + wn * 64 + j * 16 + lm;
    float bv = bias ? bias[gn] : 0.0f;
    #pragma unroll
    for (int v = 0; v < 8; ++v) {
      int gm = mbase + wm * 16 + rbase + v;
      float c = acc[j][v] + bv;
      if (do_relu) c = fmaxf(c, 0.0f);
      if (resid) c += resid[gm * N + gn];
      out[gm * N + gn] = c;
    }
  }
}

// ---------------------------------------------------------------------------
// Emit head: per-category online logsumexp over V, fused into the WMMA GEMM.
// Grid: (NSPLIT, Q_ALL/BM).  Never materializes the 2048x32000 logits.
// ---------------------------------------------------------------------------
__global__ __launch_bounds__(256)
void emit_lse_partial(const float* __restrict__ nt_emb, const float* __restrict__ emit_w,
                      const float* __restrict__ emit_b,
                      float* __restrict__ pm, float* __restrict__ ps)
{
  __shared__ float As[BM][LDP];
  __shared__ float Bs[BN][LDP];
  __shared__ float Cs[BM][BN + 4];
  __shared__ float redm[BM][4];
  __shared__ float reds[BM][4];
  const int t    = threadIdx.x;
  const int lane = t & 31;
  const int wave = t >> 5;
  const int wm   = wave >> 1;
  const int wn   = wave & 1;
  const int split = blockIdx.x;               // 0..NSPLIT-1
  const int mbase = blockIdx.y * BM;          // category block
  const int lm  = lane & 15;
  const int lk2 = (lane >> 4) * 2;
  const int rbase = (lane >> 4) * 8;

  float run_m = -3.0e38f, run_s = 0.0f;       // owned by threads t < BM (row = t)

  for (int vc = split * CHUNKS_PER_SPLIT; vc < (split + 1) * CHUNKS_PER_SPLIT; ++vc) {
    const int nbase = vc * BN;
    v8f acc[4];
    #pragma unroll
    for (int j = 0; j < 4; ++j) acc[j] = v8f_zero();

    for (int k0 = 0; k0 < DDIM; k0 += BK) {
      __syncthreads();
      #pragma unroll
      for (int i = 0; i < 2; ++i) {             // nt_emb tile: async b128
        int v = t + 256 * i;
        int r = v >> 3, c4 = (v & 7) * 4;
        async_lds_b128(&As[r][c4], &nt_emb[(mbase + r) * DDIM + k0 + c4]);
      }
      #pragma unroll
      for (int i = 0; i < 16; ++i) {            // emit_w tile, transposed store
        int idx = t + 256 * i;
        int kr = idx >> 7, nc = idx & 127;
        Bs[nc][kr] = emit_w[(k0 + kr) * VSZ + nbase + nc];
      }
      // hint the HBM stream: pull next K-chunk of emit_w toward L2
      if (k0 + BK < DDIM) {
        int kr = t >> 3, nc = (t & 7) * 16;
        __builtin_prefetch(&emit_w[(k0 + BK + kr) * VSZ + nbase + nc], 0, 3);
      }
      wait_async0();
      __syncthreads();
      #pragma unroll
      for (int kk = 0; kk < BK; kk += 4) {
        v2f a = *(const v2f*)&As[wm * 16 + lm][kk + lk2];
        #pragma unroll
        for (int j = 0; j < 4; ++j) {
          v2f b = *(const v2f*)&Bs[wn * 64 + j * 16 + lm][kk + lk2];
          acc[j] = wmma_f32(a, b, acc[j]);
        }
      }
    }

    // spill logits(+bias) tile to LDS for the row reduction
    #pragma unroll
    for (int j = 0; j < 4; ++j) {
      int ln = wn * 64 + j * 16 + lm;
      float bv = emit_b[nbase + ln];
      #pragma unroll
      for (int v = 0; v < 8; ++v)
        Cs[wm * 16 + rbase + v][ln] = acc[j][v] + bv;
    }
    __syncthreads();

    {   // 4 threads per row: partial max / sum-exp over 32 cols each
      int r = t >> 2, q4 = t & 3;
      float lmx = -3.0e38f;
      #pragma unroll
      for (int c = 0; c < 32; ++c) lmx = fmaxf(lmx, Cs[r][q4 * 32 + c]);
      float lsm = 0.0f;
      #pragma unroll
      for (int c = 0; c < 32; ++c) lsm += __expf(Cs[r][q4 * 32 + c] - lmx);
      redm[r][q4] = lmx;
      reds[r][q4] = lsm;
    }
    __syncthreads();
    if (t < BM) {       // merge 4 partials, then fold into the running state
      float cm = redm[t][0], cs = reds[t][0];
      #pragma unroll
      for (int i = 1; i < 4; ++i) {
        float m2 = redm[t][i], s2 = reds[t][i];
        float nm = fmaxf(cm, m2);
        cs = cs * __expf(cm - nm) + s2 * __expf(m2 - nm);
        cm = nm;
      }
      float nm = fmaxf(run_m, cm);
      run_s = run_s * __expf(run_m - nm) + cs * __expf(cm - nm);
      run_m = nm;
    }
    __syncthreads();
  }

  if (t < BM) {
    int q = mbase + t;
    pm[q * NSPLIT + split] = run_m;
    ps[q * NSPLIT + split] = run_s;
  }
}

__global__ __launch_bounds__(256)
void lse_combine(const float* __restrict__ pm, const float* __restrict__ ps,
                 float* __restrict__ lse)
{
  int q = blockIdx.x * 256 + threadIdx.x;
  if (q >= Q_ALL) return;
  float m = -3.0e38f;
  #pragma unroll
  for (int i = 0; i < NSPLIT; ++i) m = fmaxf(m, pm[q * NSPLIT + i]);
  float s = 0.0f;
  #pragma unroll
  for (int i = 0; i < NSPLIT; ++i) s += ps[q * NSPLIT + i] * __expf(pm[q * NSPLIT + i] - m);
  lse[q] = m + logf(s);
}

// ---------------------------------------------------------------------------
// x_scores: C[token, q] = emit_w[:, x[token]] . nt_emb[q] + emit_b[x] - lse[q]
// Gathered-column WMMA GEMM, output directly in [B,S,Q_ALL] layout.
// ---------------------------------------------------------------------------
__global__ __launch_bounds__(256)
void xscore_gemm(const int* __restrict__ x, const float* __restrict__ nt_emb,
                 const float* __restrict__ emit_w, const float* __restrict__ emit_b,
                 const float* __restrict__ lse, float* __restrict__ out)
{
  __shared__ float As[BM][LDP];
  __shared__ float Bs[BN][LDP];
  __shared__ int   Xs[BM];
  const int t    = threadIdx.x;
  const int lane = t & 31;
  const int wave = t >> 5;
  const int wm   = wave >> 1;
  const int wn   = wave & 1;
  const int tbase = blockIdx.y * BM;     // tokens
  const int qbase = blockIdx.x * BN;     // categories
  const int lm  = lane & 15;
  const int lk2 = (lane >> 4) * 2;

  if (t < BM) Xs[t] = x[tbase + t];
  __syncthreads();

  v8f acc[4];
  #pragma unroll
  for (int j = 0; j < 4; ++j) acc[j] = v8f_zero();

  for (int k0 = 0; k0 < DDIM; k0 += BK) {
    __syncthreads();
    #pragma unroll
    for (int i = 0; i < 8; ++i) {        // gather A: async b32 per element
      int idx = t + 256 * i;
      int m = idx & 63, k = idx >> 6;
      async_lds_b32(&As[m][k], &emit_w[(k0 + k) * VSZ + Xs[m]]);
    }
    #pragma unroll
    for (int i = 0; i < 4; ++i) {        // B: Bs[n][k] = nt_emb rows, async b128
      int v = t + 256 * i;
      int n = v >> 3, k4 = (v & 7) * 4;
      async_lds_b128(&Bs[n][k4], &nt_emb[(qbase + n) * DDIM + k0 + k4]);
    }
    wait_async0();
    __syncthreads();
    #pragma unroll
    for (int kk = 0; kk < BK; kk += 4) {
      v2f a = *(const v2f*)&As[wm * 16 + lm][kk + lk2];
      #pragma unroll
      for (int j = 0; j < 4; ++j) {
        v2f b = *(const v2f*)&Bs[wn * 64 + j * 16 + lm][kk + lk2];
        acc[j] = wmma_f32(a, b, acc[j]);
      }
    }
  }

  const int rbase = (lane >> 4) * 8;
  #pragma unroll
  for (int j = 0; j < 4; ++j) {
    int gq = qbase + wn * 64 + j * 16 + lm;
    float l = lse[gq];
    #pragma unroll
    for (int v = 0; v < 8; ++v) {
      int mloc = wm * 16 + rbase + v;
      out[(tbase + mloc) * Q_ALL + gq] = acc[j][v] + emit_b[Xs[mloc]] - l;
    }
  }
}

// ---------------------------------------------------------------------------
// Small kernels
// ---------------------------------------------------------------------------
__global__ __launch_bounds__(256)
void root_kernel(const float* __restrict__ mask, const float* __restrict__ w,
                 const float* __restrict__ b, float* __restrict__ out)
{
  __shared__ float red[256];
  int t = threadIdx.x;
  float lm = -3.0e38f;
  for (int i = 0; i < 8; ++i) { int c = t + 256 * i; lm = fmaxf(lm, mask[c] + w[c] + b[c]); }
  red[t] = lm; __syncthreads();
  for (int s = 128; s > 0; s >>= 1) { if (t < s) red[t] = fmaxf(red[t], red[t + s]); __syncthreads(); }
  float M = red[0]; __syncthreads();
  float ls = 0.0f;
  for (int i = 0; i < 8; ++i) { int c = t + 256 * i; ls += __expf(mask[c] + w[c] + b[c] - M); }
  red[t] = ls; __syncthreads();
  for (int s = 128; s > 0; s >>= 1) { if (t < s) red[t] += red[t + s]; __syncthreads(); }
  float L = M + logf(red[0]);
  for (int i = 0; i < 8; ++i) { int c = t + 256 * i; out[c] = mask[c] + w[c] + b[c] - L; }
}

__global__ __launch_bounds__(256)
void rule_kernel(const float* __restrict__ rule_w, const float* __restrict__ rule_b,
                 const float* __restrict__ split_scores,
                 float* __restrict__ gl, float* __restrict__ gr)
{
  __shared__ float red[256];
  int r = blockIdx.x, t = threadIdx.x;
  const float* row = rule_w + (size_t)r * (2 * Q_ARG);
  float lm = -3.0e38f;
  for (int i = 0; i < 8; ++i) { int c = t + 256 * i; lm = fmaxf(lm, row[c] + rule_b[c]); }
  red[t] = lm; __syncthreads();
  for (int s = 128; s > 0; s >>= 1) { if (t < s) red[t] = fmaxf(red[t], red[t + s]); __syncthreads(); }
  float M = red[0]; __syncthreads();
  float ls = 0.0f;
  for (int i = 0; i < 8; ++i) { int c = t + 256 * i; ls += __expf(row[c] + rule_b[c] - M); }
  red[t] = ls; __syncthreads();
  for (int s = 128; s > 0; s >>= 1) { if (t < s) red[t] += red[t + s]; __syncthreads(); }
  float L = M + logf(red[0]);
  float branch = split_scores[r * 2 + 0];
  for (int i = 0; i < 8; ++i) {
    int c = t + 256 * i;
    float v = row[c] + rule_b[c] - L + branch;
    if (c < Q_ARG) gl[(size_t)r * Q_ARG + c] = v;
    else           gr[(size_t)r * Q_ARG + (c - Q_ARG)] = v;
  }
}

__global__ __launch_bounds__(256)
void split_kernel(const float* __restrict__ h, const float* __restrict__ sow,
                  const float* __restrict__ sob, float* __restrict__ out)
{
  int wave = threadIdx.x >> 5, lane = threadIdx.x & 31;
  int row = blockIdx.x * 8 + wave;
  const float* hr = h + (size_t)row * DDIM;
  float d0 = 0.0f, d1 = 0.0f;
  for (int k = lane; k < DDIM; k += 32) {
    float hv = hr[k];
    d0 += hv * sow[k * 2 + 0];
    d1 += hv * sow[k * 2 + 1];
  }
  for (int m = 16; m > 0; m >>= 1) { d0 += __shfl_xor(d0, m, 32); d1 += __shfl_xor(d1, m, 32); }
  if (lane == 0) {
    float v0 = d0 + sob[0], v1 = d1 + sob[1];
    float M = fmaxf(v0, v1);
    float L = M + logf(__expf(v0 - M) + __expf(v1 - M));
    out[row * 2 + 0] = v0 - L;
    out[row * 2 + 1] = v1 - L;
  }
}

// ---------------------------------------------------------------------------
extern "C" void kernel_launch(void* const* d_in, const int* in_sizes, int n_in,
                              void* d_out, int out_size, void* d_ws, size_t ws_size,
                              hipStream_t stream)
{
  (void)in_sizes; (void)n_in; (void)out_size; (void)ws_size;
  const int*   x         = (const int*)  d_in[0];
  const float* nt_emb    = (const float*)d_in[1];
  const float* root_w    = (const float*)d_in[2];
  const float* root_b    = (const float*)d_in[3];
  const float* rule_w    = (const float*)d_in[4];
  const float* rule_b    = (const float*)d_in[5];
  const float* sw1       = (const float*)d_in[6];
  const float* sb1       = (const float*)d_in[7];
  const float* r1w1      = (const float*)d_in[8];
  const float* r1b1      = (const float*)d_in[9];
  const float* r1w2      = (const float*)d_in[10];
  const float* r1b2      = (const float*)d_in[11];
  const float* r2w1      = (const float*)d_in[12];
  const float* r2b1      = (const float*)d_in[13];
  const float* r2w2      = (const float*)d_in[14];
  const float* r2b2      = (const float*)d_in[15];
  const float* sow       = (const float*)d_in[16];
  const float* sob       = (const float*)d_in[17];
  const float* emit_w    = (const float*)d_in[18];
  const float* emit_b    = (const float*)d_in[19];
  const float* root_mask = (const float*)d_in[20];

  float* out    = (float*)d_out;
  float* out_p0 = out;                                  // [2048]
  float* out_gl = out_p0 + Q_ALL;                       // [1024,1024]
  float* out_gr = out_gl + (size_t)Q_RES * Q_ARG;       // [1024,1024]
  float* out_ss = out_gr + (size_t)Q_RES * Q_ARG;       // [2048,2]
  float* out_xs = out_ss + Q_ALL * 2;                   // [4096,2048]

  // Big MLP scratch lives in the x_scores region (24 MB of 33.5 MB); it is
  // fully overwritten by xscore_gemm at the end, so d_out stays deterministic.
  float* B0 = out_xs;
  float* B1 = B0 + (size_t)Q_ALL * DDIM;
  float* B2 = B1 + (size_t)Q_ALL * DDIM;
  // Small scratch in d_ws: logsumexp partials + final lse (~172 KB).
  float* pm  = (float*)d_ws;
  float* ps  = pm + Q_ALL * NSPLIT;
  float* lse = ps + Q_ALL * NSPLIT;

  dim3 blk(256);
  // root scores
  root_kernel<<<1, blk, 0, stream>>>(root_mask, root_w, root_b, out_p0);

  // split MLP: h0 = nt_emb@sw1+sb1; two residual blocks
  dim3 g_mlp(DDIM / BN, Q_ALL / BM);                    // (8, 32)
  gemm_mlp<<<g_mlp, blk, 0, stream>>>(nt_emb, sw1,  sb1,  nullptr, B0, Q_ALL, DDIM, DDIM, 0);
  gemm_mlp<<<g_mlp, blk, 0, stream>>>(B0,     r1w1, r1b1, nullptr, B1, Q_ALL, DDIM, DDIM, 1);
  gemm_mlp<<<g_mlp, blk, 0, stream>>>(B1,     r1w2, r1b2, B0,      B2, Q_ALL, DDIM, DDIM, 1);
  gemm_mlp<<<g_mlp, blk, 0, stream>>>(B2,     r2w1, r2b1, nullptr, B1, Q_ALL, DDIM, DDIM, 1);
  gemm_mlp<<<g_mlp, blk, 0, stream>>>(B1,     r2w2, r2b2, B2,      B0, Q_ALL, DDIM, DDIM, 1);

  // split scores (2-way log_softmax), then rules (+branch)
  split_kernel<<<Q_ALL / 8, blk, 0, stream>>>(B0, sow, sob, out_ss);
  rule_kernel<<<Q_RES, blk, 0, stream>>>(rule_w, rule_b, out_ss, out_gl, out_gr);

  // emit head: fused online logsumexp over V (no 262 MB logits buffer)
  dim3 g_emit(NSPLIT, Q_ALL / BM);                      // (10, 32) = 320 WGs
  emit_lse_partial<<<g_emit, blk, 0, stream>>>(nt_emb, emit_w, emit_b, pm, ps);
  lse_combine<<<Q_ALL / 256, blk, 0, stream>>>(pm, ps, lse);

  // gathered-column GEMM -> x_scores (overwrites the scratch region last)
  dim3 g_xs(Q_ALL / BN, NTOK / BM);                     // (16, 64)
  xscore_gemm<<<g_xs, blk, 0, stream>>>(x, nt_emb, emit_w, emit_b, lse, out_xs);
}